// messNet_63350767616428
// MI455X (gfx1250) — compile-verified
//
#include <hip/hip_runtime.h>

typedef __attribute__((ext_vector_type(2))) float v2f;
typedef __attribute__((ext_vector_type(4))) float v4f;
typedef __attribute__((ext_vector_type(8))) float v8f;
typedef __attribute__((ext_vector_type(4))) int   v4i;

// ---------------------------------------------------------------------------
// Element of the 4x4 augmented affine matrix [[W, b], [0, 1]], zero-padded to
// the 16-wide WMMA operand space.
// ---------------------------------------------------------------------------
__device__ __forceinline__ float affElem(const float* W, const float* b, int r, int c) {
    if (r == 3) return (c == 3) ? 1.0f : 0.0f;   // bottom affine row
    if (r > 3 || c > 3) return 0.0f;             // padding
    return (c == 3) ? b[r] : W[r * 3 + c];
}

// ---------------------------------------------------------------------------
// Single-wave kernel: Mc = M4 * M2 * M1 via two V_WMMA_F32_16X16X4_F32 ops.
// Writes 12 floats: Wc (3x3 row-major) then bc (3).
// A layout (16x4 f32): lanes 0-15 hold K=0,1 in v[0],v[1]; lanes 16-31 K=2,3.
// B layout (4x16 f32): v[0] = rows {0,2}, v[1] = rows {1,3}, cols = lane%16.
// D layout (16x16 f32): VGPR r, lanes 0-15 -> row r (r<8), col = lane.
// ---------------------------------------------------------------------------
__global__ void fuse_affine_wmma(const float* __restrict__ W1, const float* __restrict__ b1,
                                 const float* __restrict__ W2, const float* __restrict__ b2,
                                 const float* __restrict__ W4, const float* __restrict__ b4,
                                 float* __restrict__ fused) {
    const int lane = threadIdx.x;        // launched with exactly 32 threads (wave32)
    const bool hi  = lane >= 16;
    const int  m   = hi ? lane - 16 : lane;

    // A = M2, B = M1  ->  T = M2 * M1
    v2f a, b;
    a[0] = affElem(W2, b2, m, hi ? 2 : 0);
    a[1] = affElem(W2, b2, m, hi ? 3 : 1);
    b[0] = affElem(W1, b1, hi ? 2 : 0, m);
    b[1] = affElem(W1, b1, hi ? 3 : 1, m);
    v8f c = {};
    v8f t = __builtin_amdgcn_wmma_f32_16x16x4_f32(false, a, false, b,
                                                  (short)0, c, false, false);

    // Convert rows 0..3 of T (D layout) into B layout for the second multiply.
    float t2 = __shfl(t[2], m, 32);      // T[2][m] (from lanes 0-15)
    float t3 = __shfl(t[3], m, 32);      // T[3][m]
    v2f bb, a4;
    bb[0] = hi ? t2 : t[0];
    bb[1] = hi ? t3 : t[1];
    a4[0] = affElem(W4, b4, m, hi ? 2 : 0);
    a4[1] = affElem(W4, b4, m, hi ? 3 : 1);
    v8f d = __builtin_amdgcn_wmma_f32_16x16x4_f32(false, a4, false, bb,
                                                  (short)0, c, false, false);

    // Mc[r][col] = d[r] at lane col. Wc = Mc[0:3,0:3], bc = Mc[0:3,3].
    if (lane < 3) {
        fused[0 * 3 + lane] = d[0];
        fused[1 * 3 + lane] = d[1];
        fused[2 * 3 + lane] = d[2];
    } else if (lane == 3) {
        fused[9 + 0] = d[0];
        fused[9 + 1] = d[1];
        fused[9 + 2] = d[2];
    }
}

// ---------------------------------------------------------------------------
// Zero the [S][4] accumulator (sum0,sum1,sum2,count) each call.
// Regular-temporal stores: we WANT these lines resident in L2 for the atomics.
// ---------------------------------------------------------------------------
__global__ void zero_acc(v4f* __restrict__ acc, int n4) {
    int i = blockIdx.x * blockDim.x + threadIdx.x;
    if (i < n4) acc[i] = (v4f){0.0f, 0.0f, 0.0f, 0.0f};
}

// ---------------------------------------------------------------------------
// Main streaming kernel: 4 nodes per thread.
//   loads:   3x v4f (coor) + 2x v4i (edges), NON-TEMPORAL (335 MB streamed
//            once -- keep it from evicting the 16 MB L2-resident accumulator)
//   math:    fused 3x3 affine (9 fma per node, bias as fma addend)
//   scatter: 4x non-returning global_atomic_add_f32 per node (RT, L2-resident)
// ---------------------------------------------------------------------------
__device__ __forceinline__ void xform_scatter_one(
    float x, float y, float z, int idx,
    float w00, float w01, float w02,
    float w10, float w11, float w12,
    float w20, float w21, float w22,
    float bc0, float bc1, float bc2,
    float* __restrict__ acc) {
    float h0 = fmaf(w00, x, fmaf(w01, y, fmaf(w02, z, bc0)));
    float h1 = fmaf(w10, x, fmaf(w11, y, fmaf(w12, z, bc1)));
    float h2 = fmaf(w20, x, fmaf(w21, y, fmaf(w22, z, bc2)));
    float* p = acc + ((long long)idx << 2);
    (void)__hip_atomic_fetch_add(p + 0, h0,   __ATOMIC_RELAXED, __HIP_MEMORY_SCOPE_AGENT);
    (void)__hip_atomic_fetch_add(p + 1, h1,   __ATOMIC_RELAXED, __HIP_MEMORY_SCOPE_AGENT);
    (void)__hip_atomic_fetch_add(p + 2, h2,   __ATOMIC_RELAXED, __HIP_MEMORY_SCOPE_AGENT);
    (void)__hip_atomic_fetch_add(p + 3, 1.0f, __ATOMIC_RELAXED, __HIP_MEMORY_SCOPE_AGENT);
}

__global__ void transform_scatter(const v4f* __restrict__ coor4,
                                  const v4i* __restrict__ edges4,
                                  const float* __restrict__ fused,
                                  float* __restrict__ acc,
                                  int ngroups) {
    int t = blockIdx.x * blockDim.x + threadIdx.x;
    if (t >= ngroups) return;

    // Uniform params -> scalar loads through K$
    const float w00 = fused[0], w01 = fused[1], w02 = fused[2];
    const float w10 = fused[3], w11 = fused[4], w12 = fused[5];
    const float w20 = fused[6], w21 = fused[7], w22 = fused[8];
    const float bc0 = fused[9], bc1 = fused[10], bc2 = fused[11];

    v4f c0 = __builtin_nontemporal_load(coor4 + 3 * t + 0);
    v4f c1 = __builtin_nontemporal_load(coor4 + 3 * t + 1);
    v4f c2 = __builtin_nontemporal_load(coor4 + 3 * t + 2);
    v4i e0 = __builtin_nontemporal_load(edges4 + 2 * t + 0);  // nodes 4t,4t+1: idx .y,.w
    v4i e1 = __builtin_nontemporal_load(edges4 + 2 * t + 1);  // nodes 4t+2,4t+3

    xform_scatter_one(c0.x, c0.y, c0.z, e0.y, w00,w01,w02,w10,w11,w12,w20,w21,w22,bc0,bc1,bc2, acc);
    xform_scatter_one(c0.w, c1.x, c1.y, e0.w, w00,w01,w02,w10,w11,w12,w20,w21,w22,bc0,bc1,bc2, acc);
    xform_scatter_one(c1.z, c1.w, c2.x, e1.y, w00,w01,w02,w10,w11,w12,w20,w21,w22,bc0,bc1,bc2, acc);
    xform_scatter_one(c2.y, c2.z, c2.w, e1.w, w00,w01,w02,w10,w11,w12,w20,w21,w22,bc0,bc1,bc2, acc);
}

// ---------------------------------------------------------------------------
// mean = sums / max(count, 1); 4 segments per thread.
// Accumulator reads are last-use and output is write-once -> non-temporal.
// Output layout: 4 segments * 3 floats = 3x v4f per thread (16B aligned).
// ---------------------------------------------------------------------------
__global__ void finalize_mean(const v4f* __restrict__ acc,
                              v4f* __restrict__ out, int ngroups) {
    int t = blockIdx.x * blockDim.x + threadIdx.x;
    if (t >= ngroups) return;

    v4f a0 = __builtin_nontemporal_load(acc + 4 * t + 0);
    v4f a1 = __builtin_nontemporal_load(acc + 4 * t + 1);
    v4f a2 = __builtin_nontemporal_load(acc + 4 * t + 2);
    v4f a3 = __builtin_nontemporal_load(acc + 4 * t + 3);

    float i0 = 1.0f / fmaxf(a0.w, 1.0f);
    float i1 = 1.0f / fmaxf(a1.w, 1.0f);
    float i2 = 1.0f / fmaxf(a2.w, 1.0f);
    float i3 = 1.0f / fmaxf(a3.w, 1.0f);

    v4f o0 = {a0.x * i0, a0.y * i0, a0.z * i0, a1.x * i1};
    v4f o1 = {a1.y * i1, a1.z * i1, a2.x * i2, a2.y * i2};
    v4f o2 = {a2.z * i2, a3.x * i3, a3.y * i3, a3.z * i3};

    __builtin_nontemporal_store(o0, out + 3 * t + 0);
    __builtin_nontemporal_store(o1, out + 3 * t + 1);
    __builtin_nontemporal_store(o2, out + 3 * t + 2);
}

extern "C" void kernel_launch(void* const* d_in, const int* in_sizes, int n_in,
                              void* d_out, int out_size, void* d_ws, size_t ws_size,
                              hipStream_t stream) {
    const int*   edges = (const int*)d_in[0];
    const float* coor  = (const float*)d_in[1];
    const float* W1 = (const float*)d_in[2];
    const float* b1 = (const float*)d_in[3];
    const float* W2 = (const float*)d_in[4];
    const float* b2 = (const float*)d_in[5];
    const float* W4 = (const float*)d_in[6];
    const float* b4 = (const float*)d_in[7];

    const int N = in_sizes[0] / 2;     // edges is [1,N,2]
    const int S = out_size / 3;        // out is [1,S,3]

    float* fused = (float*)d_ws;            // 12 floats (64B-aligned slot)
    float* acc   = (float*)d_ws + 16;       // [S][4] floats = 16 MB

    const int TB = 256;

    zero_acc<<<(S + TB - 1) / TB, TB, 0, stream>>>((v4f*)acc, S);

    fuse_affine_wmma<<<1, 32, 0, stream>>>(W1, b1, W2, b2, W4, b4, fused);

    const int ngroups = N / 4;
    transform_scatter<<<(ngroups + TB - 1) / TB, TB, 0, stream>>>(
        (const v4f*)coor, (const v4i*)edges, fused, acc, ngroups);

    const int fgroups = S / 4;
    finalize_mean<<<(fgroups + TB - 1) / TB, TB, 0, stream>>>(
        (const v4f*)acc, (v4f*)d_out, fgroups);
}